// HeterogeneousGNN_7490422964797
// MI455X (gfx1250) — compile-verified
//
#include <hip/hip_runtime.h>
#include <hip/hip_bf16.h>

#define NUM_OPS 100000
#define NUM_MACH 512
#define HDIM 128
#define OPF 6
#define MF 2
#define NLAYERS 2
#define N_PREC 800000
#define N_COMPAT 1600000

typedef __attribute__((ext_vector_type(2))) float v2f;
typedef __attribute__((ext_vector_type(8))) float v8f;

// ---------------------------------------------------------------------------
// Dense embedding: out[node,h] = sum_f F[node,f]*W[h,f] + b[h]
// ---------------------------------------------------------------------------
__global__ __launch_bounds__(256)
void embed_kernel(const float* __restrict__ F, const float* __restrict__ W,
                  const float* __restrict__ bias, float* __restrict__ out,
                  int n, int nf) {
    int idx = blockIdx.x * blockDim.x + threadIdx.x;
    if (idx >= n * HDIM) return;
    int node = idx >> 7;       // / 128
    int h    = idx & 127;
    const float* f = F + (size_t)node * nf;
    const float* w = W + (size_t)h * nf;
    float acc = bias[h];
    for (int j = 0; j < nf; ++j) acc += f[j] * w[j];
    out[idx] = acc;
}

// ---------------------------------------------------------------------------
// Y[N,128] = X[N,128] @ W[128,128]^T + b   via V_WMMA_F32_16X16X4_F32.
// Block = 256 threads (8 waves). Block handles a 16-row M-tile; wave w handles
// output columns [16w, 16w+16). X tile staged in LDS (8KB), W rows from L2.
//
// Per ISA 7.12.2:
//   A (16x4 f32):  lanes 0-15 -> M=lane,  VGPR0/1 = K0/K1
//                  lanes 16-31-> M=lane-16,VGPR0/1 = K2/K3
//   B (4x16):      mirrored (lanes 0-15: N=lane, K0/K1; 16-31: K2/K3)
//   C/D (16x16):   VGPR v: lanes 0-15 -> M=v, N=lane; lanes 16-31 -> M=v+8
// ---------------------------------------------------------------------------
__global__ __launch_bounds__(256)
void gemm_xWt_bias(const float* __restrict__ X, const float* __restrict__ W,
                   const float* __restrict__ bias, float* __restrict__ Y) {
    __shared__ __align__(16) float xs[16 * HDIM];   // 8 KB X tile
    const int m0  = blockIdx.x * 16;
    const int tid = threadIdx.x;

    // Stage 16x128 floats: 512 float4, 256 threads -> 2 each
    {
        const float4* src = (const float4*)(X + (size_t)m0 * HDIM);
        float4* dst = (float4*)xs;
        dst[tid]       = src[tid];
        dst[tid + 256] = src[tid + 256];
    }
    __syncthreads();

    const int wave = tid >> 5;           // 0..7 : N-tile
    const int lane = tid & 31;
    const int half = lane >> 4;          // 0 or 1 (selects K pair)
    const int l16  = lane & 15;
    const int n    = wave * 16 + l16;    // output column
    const float* wrow = W + (size_t)n * HDIM;   // row n of W == column n of W^T

    v8f c = {0.f, 0.f, 0.f, 0.f, 0.f, 0.f, 0.f, 0.f};
    #pragma unroll 4
    for (int k = 0; k < HDIM; k += 4) {
        const int ka = k + half * 2;
        v2f a, b;
        a.x = xs[l16 * HDIM + ka];
        a.y = xs[l16 * HDIM + ka + 1];
        b.x = wrow[ka];
        b.y = wrow[ka + 1];
        // 8 args: (neg_a, A, neg_b, B, c_mod, C, reuse_a, reuse_b)
        c = __builtin_amdgcn_wmma_f32_16x16x4_f32(false, a, false, b,
                                                  (short)0, c, false, false);
    }

    const float bn    = bias[n];
    const int   mbase = m0 + half * 8;
    #pragma unroll
    for (int v = 0; v < 8; ++v) {
        Y[(size_t)(mbase + v) * HDIM + n] = c[v] + bn;
    }
}

// ---------------------------------------------------------------------------
// Precedence scatter: one wave per edge (grid-stride).
// sum[tgt] += T[src]; cnt[tgt] += 1
// ---------------------------------------------------------------------------
__global__ __launch_bounds__(256)
void prec_scatter(const float* __restrict__ T, const int* __restrict__ src,
                  const int* __restrict__ tgt, float* __restrict__ sum,
                  float* __restrict__ cnt, int nedges) {
    const int lane = threadIdx.x & 31;
    int wid = (blockIdx.x * blockDim.x + threadIdx.x) >> 5;
    const int nw = (gridDim.x * blockDim.x) >> 5;
    for (int e = wid; e < nedges; e += nw) {
        const int s = src[e];
        const int t = tgt[e];
        const float* ts = T   + (size_t)s * HDIM;
        float*       ds = sum + (size_t)t * HDIM;
        #pragma unroll
        for (int j = 0; j < 4; ++j) {
            const int h = lane + j * 32;
            atomicAdd(&ds[h], ts[h]);
        }
        if (lane == 0) atomicAdd(&cnt[t], 1.0f);
    }
}

// ---------------------------------------------------------------------------
// Compat scatter (both directions in one pass over the 1.6M edges):
//   op->mach: machSum[tgt-NUM_OPS] += opC[src]
//   mach->op: opSum[tgt]           += machC[src-NUM_OPS]
// ---------------------------------------------------------------------------
__global__ __launch_bounds__(256)
void compat_scatter(const float* __restrict__ opC, const float* __restrict__ machC,
                    const int* __restrict__ src, const int* __restrict__ tgt,
                    float* __restrict__ machSum, float* __restrict__ machCnt,
                    float* __restrict__ opSum,   float* __restrict__ opCnt,
                    int nedges) {
    const int lane = threadIdx.x & 31;
    int wid = (blockIdx.x * blockDim.x + threadIdx.x) >> 5;
    const int nw = (gridDim.x * blockDim.x) >> 5;
    for (int e = wid; e < nedges; e += nw) {
        const int s = src[e];
        const int t = tgt[e];
        if (s < NUM_OPS && t >= NUM_OPS) {
            const float* g = opC    + (size_t)s * HDIM;
            float*       d = machSum + (size_t)(t - NUM_OPS) * HDIM;
            #pragma unroll
            for (int j = 0; j < 4; ++j) {
                const int h = lane + j * 32;
                atomicAdd(&d[h], g[h]);
            }
            if (lane == 0) atomicAdd(&machCnt[t - NUM_OPS], 1.0f);
        } else if (s >= NUM_OPS && t < NUM_OPS) {
            const float* g = machC + (size_t)(s - NUM_OPS) * HDIM;
            float*       d = opSum + (size_t)t * HDIM;
            #pragma unroll
            for (int j = 0; j < 4; ++j) {
                const int h = lane + j * 32;
                atomicAdd(&d[h], g[h]);
            }
            if (lane == 0) atomicAdd(&opCnt[t], 1.0f);
        }
    }
}

// ---------------------------------------------------------------------------
// emb = LayerNorm(emb + sum1/max(cnt1,1) + sum2/max(cnt2,1)) * g + b
// sum1/cnt1 may be null (machine side has only the compat aggregation).
// One 128-thread block per node; LDS tree reductions for mean/var.
// ---------------------------------------------------------------------------
__global__ __launch_bounds__(128)
void combine_ln(float* __restrict__ emb,
                const float* __restrict__ sum1, const float* __restrict__ cnt1,
                const float* __restrict__ sum2, const float* __restrict__ cnt2,
                const float* __restrict__ g, const float* __restrict__ b) {
    const int node = blockIdx.x;
    const int h = threadIdx.x;
    __shared__ float red[HDIM];

    float x = emb[(size_t)node * HDIM + h];
    if (sum1 != nullptr) {
        x += sum1[(size_t)node * HDIM + h] / fmaxf(cnt1[node], 1.0f);
    }
    x += sum2[(size_t)node * HDIM + h] / fmaxf(cnt2[node], 1.0f);

    red[h] = x;
    __syncthreads();
    for (int s = 64; s > 0; s >>= 1) { if (h < s) red[h] += red[h + s]; __syncthreads(); }
    const float mu = red[0] * (1.0f / HDIM);
    __syncthreads();

    const float d = x - mu;
    red[h] = d * d;
    __syncthreads();
    for (int s = 64; s > 0; s >>= 1) { if (h < s) red[h] += red[h + s]; __syncthreads(); }
    const float var = red[0] * (1.0f / HDIM);

    const float r = rsqrtf(var + 1e-5f);
    emb[(size_t)node * HDIM + h] = d * r * g[h] + b[h];
}

// ---------------------------------------------------------------------------
extern "C" void kernel_launch(void* const* d_in, const int* in_sizes, int n_in,
                              void* d_out, int out_size, void* d_ws, size_t ws_size,
                              hipStream_t stream) {
    const float* op_features  = (const float*)d_in[0];
    const float* mach_features = (const float*)d_in[1];
    const int*   prec_edges   = (const int*)d_in[2];   // [2, N_PREC]
    const int*   compat_edges = (const int*)d_in[3];   // [2, N_COMPAT]
    const float* op_emb_W  = (const float*)d_in[4];
    const float* op_emb_b  = (const float*)d_in[5];
    const float* mach_emb_W = (const float*)d_in[6];
    const float* mach_emb_b = (const float*)d_in[7];
    const float* prec_W    = (const float*)d_in[8];    // [L,H,H]
    const float* prec_b    = (const float*)d_in[9];    // [L,H]
    const float* compat_W  = (const float*)d_in[10];
    const float* compat_b  = (const float*)d_in[11];
    const float* op_ln_g   = (const float*)d_in[12];
    const float* op_ln_b   = (const float*)d_in[13];
    const float* mach_ln_g = (const float*)d_in[14];
    const float* mach_ln_b = (const float*)d_in[15];

    // Persistent embeddings live directly in d_out (output = final embeddings).
    float* op_emb   = (float*)d_out;                          // [NUM_OPS, H]
    float* mach_emb = (float*)d_out + (size_t)NUM_OPS * HDIM; // [NUM_MACH, H]

    // Workspace carve-up
    float* ws = (float*)d_ws;
    float* T        = ws;  ws += (size_t)NUM_OPS * HDIM;   // transformed ops (reused prec->compat)
    float* precSum  = ws;  ws += (size_t)NUM_OPS * HDIM;
    float* opCSum   = ws;  ws += (size_t)NUM_OPS * HDIM;
    float* precCnt  = ws;  ws += NUM_OPS;
    float* opCCnt   = ws;  ws += NUM_OPS;
    float* machC    = ws;  ws += (size_t)NUM_MACH * HDIM;
    float* machSum  = ws;  ws += (size_t)NUM_MACH * HDIM;
    float* machCnt  = ws;  ws += NUM_MACH;

    const int* p_src = prec_edges;
    const int* p_tgt = prec_edges + N_PREC;
    const int* c_src = compat_edges;
    const int* c_tgt = compat_edges + N_COMPAT;

    // Initial embeddings
    embed_kernel<<<(NUM_OPS * HDIM + 255) / 256, 256, 0, stream>>>(
        op_features, op_emb_W, op_emb_b, op_emb, NUM_OPS, OPF);
    embed_kernel<<<(NUM_MACH * HDIM + 255) / 256, 256, 0, stream>>>(
        mach_features, mach_emb_W, mach_emb_b, mach_emb, NUM_MACH, MF);

    for (int l = 0; l < NLAYERS; ++l) {
        const float* pW = prec_W   + (size_t)l * HDIM * HDIM;
        const float* pb = prec_b   + (size_t)l * HDIM;
        const float* cW = compat_W + (size_t)l * HDIM * HDIM;
        const float* cb = compat_b + (size_t)l * HDIM;

        // ---- precedence message: T = op_emb @ prec_W^T + prec_b
        gemm_xWt_bias<<<NUM_OPS / 16, 256, 0, stream>>>(op_emb, pW, pb, T);
        hipMemsetAsync(precSum, 0, (size_t)NUM_OPS * HDIM * sizeof(float), stream);
        hipMemsetAsync(precCnt, 0, (size_t)NUM_OPS * sizeof(float), stream);
        prec_scatter<<<2048, 256, 0, stream>>>(T, p_src, p_tgt, precSum, precCnt, N_PREC);

        // ---- compat messages (T reused for op_c; stream order makes this safe)
        gemm_xWt_bias<<<NUM_OPS / 16, 256, 0, stream>>>(op_emb, cW, cb, T);
        gemm_xWt_bias<<<NUM_MACH / 16, 256, 0, stream>>>(mach_emb, cW, cb, machC);
        hipMemsetAsync(opCSum, 0, (size_t)NUM_OPS * HDIM * sizeof(float), stream);
        hipMemsetAsync(opCCnt, 0, (size_t)NUM_OPS * sizeof(float), stream);
        hipMemsetAsync(machSum, 0, (size_t)NUM_MACH * HDIM * sizeof(float), stream);
        hipMemsetAsync(machCnt, 0, (size_t)NUM_MACH * sizeof(float), stream);
        compat_scatter<<<4096, 256, 0, stream>>>(T, machC, c_src, c_tgt,
                                                 machSum, machCnt, opCSum, opCCnt,
                                                 N_COMPAT);

        // ---- residual + LayerNorm
        combine_ln<<<NUM_OPS, 128, 0, stream>>>(op_emb, precSum, precCnt,
                                                opCSum, opCCnt,
                                                op_ln_g + (size_t)l * HDIM,
                                                op_ln_b + (size_t)l * HDIM);
        combine_ln<<<NUM_MACH, 128, 0, stream>>>(mach_emb, nullptr, nullptr,
                                                 machSum, machCnt,
                                                 mach_ln_g + (size_t)l * HDIM,
                                                 mach_ln_b + (size_t)l * HDIM);
    }
}